// GraphClassifier_34617436406358
// MI455X (gfx1250) — compile-verified
//
#include <hip/hip_runtime.h>
#include <hip/hip_bf16.h>
#include <math.h>

#define N_NODES  50000
#define N_EDGES  800000
#define HID      128
#define N_GRAPHS 256

typedef __attribute__((ext_vector_type(2))) float v2f;
typedef __attribute__((ext_vector_type(8))) float v8f;

// ---------------------------------------------------------------------------
// Generic fp32 WMMA GEMM: C[M,N] = act(A[M,K] @ B[K,N] + bias)
// Block = 256 threads (8 waves). Block tile = 16 rows x 128 cols,
// one 16x16 output tile per wave. A tile staged in LDS (stride K+1 to kill
// bank conflicts on the 16-lane column reads). M%16==0, N%128==0, K%4==0.
// ---------------------------------------------------------------------------
__global__ __launch_bounds__(256) void gemm_wmma_f32(
    const float* __restrict__ A, const float* __restrict__ B,
    float* __restrict__ C, int M, int N, int K,
    const float* __restrict__ bias, int act)
{
    __shared__ float ldsA[16 * 257];   // supports K up to 256
    const int t    = threadIdx.x;
    const int lane = t & 31;
    const int w    = t >> 5;
    const int row0 = blockIdx.x * 16;
    const int col0 = blockIdx.y * 128 + w * 16;
    const int sA   = K + 1;

    // cooperative, coalesced load of the 16 x K A-tile
    const int total = 16 * K;
    for (int f = t; f < total; f += 256) {
        int r = f / K;
        int k = f - r * K;
        ldsA[r * sA + k] = A[(size_t)(row0 + r) * K + k];
    }
    __syncthreads();

    const int m  = lane & 15;         // A row / B col within tile
    const int kh = (lane >> 4) * 2;   // which K pair this half-wave holds

    v8f acc = {};
    for (int k0 = 0; k0 < K; k0 += 4) {
        v2f a, b;
        a.x = ldsA[m * sA + k0 + kh];
        a.y = ldsA[m * sA + k0 + kh + 1];
        const float* bp = B + (size_t)(k0 + kh) * N + col0 + m;
        b.x = bp[0];
        b.y = bp[N];
        acc = __builtin_amdgcn_wmma_f32_16x16x4_f32(
            false, a, false, b, (short)0, acc, false, false);
    }

    const float bv   = bias ? bias[col0 + m] : 0.0f;
    const int rbase  = (lane >> 4) * 8;
    for (int r = 0; r < 8; ++r) {
        float v = acc[r] + bv;
        if (act == 1) v = v > 0.0f ? v : 0.0f;   // ReLU
        C[(size_t)(row0 + rbase + r) * N + col0 + m] = v;
    }
}

// ---------------------------------------------------------------------------
// e_src[i] = h[i,:] . a[0:128]   e_dst[i] = h[i,:] . a[128:256]
// one wave per node, lane owns 4 features, wave32 xor-shuffle reduction
// ---------------------------------------------------------------------------
__global__ void attn_coef(const float* __restrict__ h, const float* __restrict__ a,
                          float* __restrict__ e_src, float* __restrict__ e_dst,
                          int nNodes)
{
    int node = (blockIdx.x * blockDim.x + threadIdx.x) >> 5;
    int lane = threadIdx.x & 31;
    if (node >= nNodes) return;

    const float4 hv  = *reinterpret_cast<const float4*>(h + (size_t)node * HID + lane * 4);
    const float4 alo = *reinterpret_cast<const float4*>(a + lane * 4);
    const float4 ahi = *reinterpret_cast<const float4*>(a + HID + lane * 4);

    float s1 = hv.x * alo.x + hv.y * alo.y + hv.z * alo.z + hv.w * alo.w;
    float s2 = hv.x * ahi.x + hv.y * ahi.y + hv.z * ahi.z + hv.w * ahi.w;
    for (int off = 16; off > 0; off >>= 1) {
        s1 += __shfl_xor(s1, off, 32);
        s2 += __shfl_xor(s2, off, 32);
    }
    if (lane == 0) { e_src[node] = s1; e_dst[node] = s2; }
}

// ---------------------------------------------------------------------------
// Per edge: ee = exp(-leaky_relu(e_src[s]+e_dst[d], 0.2))
//   rowsum[s] += ee ;  num[s,:] += ee * h[d,:]
// one wave per edge, lane owns 4 features; L2-resident gathers + f32 atomics
// ---------------------------------------------------------------------------
__global__ void edge_aggregate(const int* __restrict__ src, const int* __restrict__ dst,
                               const float* __restrict__ e_src, const float* __restrict__ e_dst,
                               const float* __restrict__ h,
                               float* __restrict__ rowsum, float* __restrict__ num,
                               int nE)
{
    int e    = (blockIdx.x * blockDim.x + threadIdx.x) >> 5;
    int lane = threadIdx.x & 31;
    if (e >= nE) return;

    int s = src[e];
    int d = dst[e];
    float ev = e_src[s] + e_dst[d];
    float lr = ev > 0.0f ? ev : 0.2f * ev;
    float ee = expf(-lr);

    if (lane == 0) atomicAdd(rowsum + s, ee);

    const float4 hv = *reinterpret_cast<const float4*>(h + (size_t)d * HID + lane * 4);
    float* np = num + (size_t)s * HID + lane * 4;
    atomicAdd(np + 0, ee * hv.x);
    atomicAdd(np + 1, ee * hv.y);
    atomicAdd(np + 2, ee * hv.z);
    atomicAdd(np + 3, ee * hv.w);
}

// out = elu(num / (rowsum + 1e-16))
__global__ void gat_finalize(const float* __restrict__ num, const float* __restrict__ rowsum,
                             float* __restrict__ out, int total)
{
    int stride = gridDim.x * blockDim.x;
    for (int i = blockIdx.x * blockDim.x + threadIdx.x; i < total; i += stride) {
        int row = i >> 7;
        float v = num[i] / (rowsum[row] + 1e-16f);
        out[i] = v > 0.0f ? v : expm1f(v);
    }
}

// ---------------------------------------------------------------------------
// BatchNorm (training mode, biased variance) over the node dimension
// ---------------------------------------------------------------------------
__global__ void bn_partial(const float* __restrict__ x,
                           float* __restrict__ sums, float* __restrict__ sumsq,
                           int nNodes)
{
    int col = threadIdx.x;                 // blockDim.x == 128
    float s = 0.0f, q = 0.0f;
    for (int r = blockIdx.x; r < nNodes; r += gridDim.x) {
        float v = x[(size_t)r * HID + col];
        s += v;
        q += v * v;
    }
    atomicAdd(sums + col, s);
    atomicAdd(sumsq + col, q);
}

__global__ void bn_stats(const float* __restrict__ sums, const float* __restrict__ sumsq,
                         float* __restrict__ mean, float* __restrict__ istd, float n)
{
    int c = threadIdx.x;                   // 128 threads
    float mu  = sums[c] / n;
    float var = sumsq[c] / n - mu * mu;
    mean[c] = mu;
    istd[c] = rsqrtf(var + 1e-5f);
}

__global__ void bn_apply(float* __restrict__ x,
                         const float* __restrict__ mean, const float* __restrict__ istd,
                         const float* __restrict__ g, const float* __restrict__ b,
                         int total)
{
    int stride = gridDim.x * blockDim.x;
    for (int i = blockIdx.x * blockDim.x + threadIdx.x; i < total; i += stride) {
        int c = i & (HID - 1);
        x[i] = (x[i] - mean[c]) * istd[c] * g[c] + b[c];
    }
}

// per-graph sum pooling: pooled[gi[i],:] += x[i,:]   (wave per node)
__global__ void pool_kernel(const float* __restrict__ x, const int* __restrict__ gi,
                            float* __restrict__ pooled, int nNodes)
{
    int node = (blockIdx.x * blockDim.x + threadIdx.x) >> 5;
    int lane = threadIdx.x & 31;
    if (node >= nNodes) return;
    int g = gi[node];
    const float4 xv = *reinterpret_cast<const float4*>(x + (size_t)node * HID + lane * 4);
    float* pp = pooled + (size_t)g * HID + lane * 4;
    atomicAdd(pp + 0, xv.x);
    atomicAdd(pp + 1, xv.y);
    atomicAdd(pp + 2, xv.z);
    atomicAdd(pp + 3, xv.w);
}

// logits[g,c] = x2[g,:] . fc3_w[:,c] + fc3_b[c]   (256x2, trivial)
__global__ void fc3_kernel(const float* __restrict__ x, const float* __restrict__ w,
                           const float* __restrict__ b, float* __restrict__ out)
{
    int idx = blockIdx.x * blockDim.x + threadIdx.x;
    if (idx >= N_GRAPHS * 2) return;
    int g = idx >> 1, c = idx & 1;
    float s = b[c];
    for (int k = 0; k < HID; ++k) s += x[g * HID + k] * w[k * 2 + c];
    out[idx] = s;
}

__global__ void zero_kernel(float* __restrict__ p, int n)
{
    int stride = gridDim.x * blockDim.x;
    for (int i = blockIdx.x * blockDim.x + threadIdx.x; i < n; i += stride) p[i] = 0.0f;
}

// ---------------------------------------------------------------------------
extern "C" void kernel_launch(void* const* d_in, const int* in_sizes, int n_in,
                              void* d_out, int out_size, void* d_ws, size_t ws_size,
                              hipStream_t stream)
{
    (void)in_sizes; (void)n_in; (void)out_size; (void)ws_size;

    const int*   adjacency = (const int*)d_in[0];       // [2, E]
    const float* input_f   = (const float*)d_in[1];     // [N, 128]
    const int*   graph_ind = (const int*)d_in[2];       // [N]
    // d_in[3] = labels (unused by forward)
    const float* W1    = (const float*)d_in[4];
    const float* a1    = (const float*)d_in[5];
    const float* bn2_g = (const float*)d_in[6];
    const float* bn2_b = (const float*)d_in[7];
    const float* W2    = (const float*)d_in[8];
    const float* a2    = (const float*)d_in[9];
    const float* bn3_g = (const float*)d_in[10];
    const float* bn3_b = (const float*)d_in[11];
    const float* W3    = (const float*)d_in[12];
    const float* a3    = (const float*)d_in[13];
    const float* fc1_w = (const float*)d_in[14];
    const float* fc1_b = (const float*)d_in[15];
    const float* fc2_w = (const float*)d_in[16];
    const float* fc2_b = (const float*)d_in[17];
    const float* fc3_w = (const float*)d_in[18];
    const float* fc3_b = (const float*)d_in[19];

    const int* src = adjacency;
    const int* dst = adjacency + N_EDGES;

    char*  ws  = (char*)d_ws;
    size_t off = 0;
    auto carve = [&](size_t bytes) -> char* {
        char* p = ws + off;
        off += (bytes + 255) & ~(size_t)255;
        return p;
    };
    float* h      = (float*)carve((size_t)N_NODES * HID * 4);
    float* num    = (float*)carve((size_t)N_NODES * HID * 4);
    float* act    = (float*)carve((size_t)N_NODES * HID * 4);
    float* e_src  = (float*)carve((size_t)N_NODES * 4);
    float* e_dst  = (float*)carve((size_t)N_NODES * 4);
    float* rowsum = (float*)carve((size_t)N_NODES * 4);
    float* sums   = (float*)carve(HID * 4);
    float* sumsq  = (float*)carve(HID * 4);
    float* mean   = (float*)carve(HID * 4);
    float* istd   = (float*)carve(HID * 4);
    float* pooled = (float*)carve((size_t)N_GRAPHS * HID * 4);
    float* mlp1   = (float*)carve((size_t)N_GRAPHS * 2 * HID * 4);
    float* mlp2   = (float*)carve((size_t)N_GRAPHS * HID * 4);

    auto gat_layer = [&](const float* x, const float* W, const float* a) {
        zero_kernel<<<1024, 256, 0, stream>>>(num, N_NODES * HID);
        zero_kernel<<<256, 256, 0, stream>>>(rowsum, N_NODES);
        gemm_wmma_f32<<<dim3(N_NODES / 16, 1), 256, 0, stream>>>(
            x, W, h, N_NODES, HID, HID, nullptr, 0);
        attn_coef<<<(N_NODES + 7) / 8, 256, 0, stream>>>(h, a, e_src, e_dst, N_NODES);
        edge_aggregate<<<(N_EDGES + 7) / 8, 256, 0, stream>>>(
            src, dst, e_src, e_dst, h, rowsum, num, N_EDGES);
        gat_finalize<<<4096, 256, 0, stream>>>(num, rowsum, act, N_NODES * HID);
    };

    auto batchnorm = [&](const float* g, const float* b) {
        zero_kernel<<<1, 256, 0, stream>>>(sums, HID);
        zero_kernel<<<1, 256, 0, stream>>>(sumsq, HID);
        bn_partial<<<256, 128, 0, stream>>>(act, sums, sumsq, N_NODES);
        bn_stats<<<1, 128, 0, stream>>>(sums, sumsq, mean, istd, (float)N_NODES);
        bn_apply<<<4096, 256, 0, stream>>>(act, mean, istd, g, b, N_NODES * HID);
    };

    gat_layer(input_f, W1, a1);
    batchnorm(bn2_g, bn2_b);
    gat_layer(act, W2, a2);
    batchnorm(bn3_g, bn3_b);
    gat_layer(act, W3, a3);

    zero_kernel<<<128, 256, 0, stream>>>(pooled, N_GRAPHS * HID);
    pool_kernel<<<(N_NODES + 7) / 8, 256, 0, stream>>>(act, graph_ind, pooled, N_NODES);

    gemm_wmma_f32<<<dim3(N_GRAPHS / 16, 2), 256, 0, stream>>>(
        pooled, fc1_w, mlp1, N_GRAPHS, 2 * HID, HID, fc1_b, 1);
    gemm_wmma_f32<<<dim3(N_GRAPHS / 16, 1), 256, 0, stream>>>(
        mlp1, fc2_w, mlp2, N_GRAPHS, HID, 2 * HID, fc2_b, 1);
    fc3_kernel<<<2, 256, 0, stream>>>(mlp2, fc3_w, fc3_b, (float*)d_out);
}